// ConvAttentionPre_6193342841606
// MI455X (gfx1250) — compile-verified
//
#include <hip/hip_runtime.h>

typedef float v2f __attribute__((ext_vector_type(2)));
typedef float v8f __attribute__((ext_vector_type(8)));

constexpr int   B_ = 16;
constexpr int   S_ = 2048;
constexpr int   D_ = 128;
constexpr float SCALE = 0.088388347648318447f;   // 1/sqrt(128)
constexpr int   PITCH = 2052;                    // 2048 + 4: float4-aligned rows, banks rotate by 4
constexpr int   SLAB_ROWS = 18;                  // 16 output rows + 2 conv halo rows
constexpr size_t CTX_ELEMS = (size_t)B_ * S_ * D_;
constexpr size_t LDS_BYTES = (size_t)SLAB_ROWS * PITCH * sizeof(float);  // 147744 B

// ---------------------------------------------------------------------------
// Fully fused: one block per (b, 16 query rows).
//  A: score slab (18 x 2048) = Q[q0-1 .. q0+16] . K^T * scale  (WMMA f32 16x16x4)
//  B: depthwise 3x3 conv + bias + relu + residual + mask, in-place in LDS
//  C: wave-per-row softmax (shfl reductions)
//  D: atten tile store (coalesced float4)
//  E: context = P @ V (WMMA f32 16x16x4)
// No HBM scratch: the 256 MiB score tensor never leaves the WGP.
// ---------------------------------------------------------------------------
__global__ __launch_bounds__(256)
void fused_conv_attention_kernel(const float* __restrict__ Q,
                                 const float* __restrict__ K,
                                 const float* __restrict__ V,
                                 const int*   __restrict__ mask,
                                 const float* __restrict__ conv_w,
                                 const float* __restrict__ conv_b,
                                 float* __restrict__ out) {
  extern __shared__ float slab[];        // [SLAB_ROWS][PITCH]; slab[i] = score row (q0-1+i)
  const int b     = blockIdx.y;
  const int q0    = blockIdx.x * 16;
  const int tid   = threadIdx.x;
  const int wave  = tid >> 5;
  const int lane  = tid & 31;
  const int m     = lane & 15;           // A row / B column owned by this lane
  const int koff  = (lane >> 4) << 1;    // K offset 0 or 2 (f32 WMMA lane layout)
  const int rbase = (lane >> 4) * 8;     // C/D layout: VGPR r -> M = rbase + r

  // ---- Phase A: build score slab with WMMA ------------------------------
  {
    int rA0 = q0 - 1 + m;                        // rows q0-1 .. q0+14 -> slab 0..15
    rA0 = rA0 < 0 ? 0 : (rA0 >= S_ ? S_ - 1 : rA0);
    int rA1 = q0 + 15 + m;                       // rows q0+15, q0+16 -> slab 16,17 (m=0,1)
    rA1 = rA1 >= S_ ? S_ - 1 : rA1;
    const float* __restrict__ qa0 = Q + ((size_t)b * S_ + rA0) * D_;
    const float* __restrict__ qa1 = Q + ((size_t)b * S_ + rA1) * D_;

    for (int nc = 0; nc < S_ / 128; ++nc) {      // 128-column chunks; wave owns 16 cols
      const int n0c = nc * 128 + wave * 16;
      const float* __restrict__ kb = K + ((size_t)b * S_ + n0c + m) * D_;
      if (nc + 1 < S_ / 128)
        __builtin_prefetch(kb + (size_t)128 * D_, 0, 1);   // global_prefetch_b8

      v8f acc0 = {0.f, 0.f, 0.f, 0.f, 0.f, 0.f, 0.f, 0.f};
      v8f acc1 = {0.f, 0.f, 0.f, 0.f, 0.f, 0.f, 0.f, 0.f};
#pragma unroll 4
      for (int dd = 0; dd < D_; dd += 4) {
        v2f a0 = *(const v2f*)(qa0 + dd + koff);
        v2f a1 = *(const v2f*)(qa1 + dd + koff);
        v2f bb = *(const v2f*)(kb + dd + koff);
        acc0 = __builtin_amdgcn_wmma_f32_16x16x4_f32(false, a0, false, bb,
                                                     (short)0, acc0, false, false);
        acc1 = __builtin_amdgcn_wmma_f32_16x16x4_f32(false, a1, false, bb,
                                                     (short)0, acc1, false, false);
      }
#pragma unroll
      for (int r = 0; r < 8; ++r)
        slab[(size_t)(rbase + r) * PITCH + n0c + m] = acc0[r] * SCALE;
      if (rbase == 0) {                          // only rows 16,17 of tile 1 are real
#pragma unroll
        for (int r = 0; r < 2; ++r)
          slab[(size_t)(16 + r) * PITCH + n0c + m] = acc1[r] * SCALE;
      }
    }
  }
  __syncthreads();
  // Zero the conv halo rows that fall outside the score map (SAME padding).
  if (q0 == 0)
    for (int k = tid; k < S_; k += 256) slab[k] = 0.0f;
  if (q0 + 16 == S_)
    for (int k = tid; k < S_; k += 256) slab[(size_t)17 * PITCH + k] = 0.0f;
  __syncthreads();

  // ---- Phase B: conv + bias + relu + residual + mask, in-place ----------
  // conv row r reads slab[r..r+2] (score) and overwrites the dead row slab[r].
  const int*   __restrict__ mrow = mask + (size_t)b * S_;
  const float* __restrict__ w    = conv_w + b * 9;
  const float w00 = w[0], w01 = w[1], w02 = w[2];
  const float w10 = w[3], w11 = w[4], w12 = w[5];
  const float w20 = w[6], w21 = w[7], w22 = w[8];
  const float bias = conv_b[b];

  for (int r = 0; r < 16; ++r) {
    const float* rowm = slab + (size_t)r * PITCH;
    const float* rowc = slab + (size_t)(r + 1) * PITCH;
    const float* rowp = slab + (size_t)(r + 2) * PITCH;
    float vout[8];
#pragma unroll
    for (int c = 0; c < 8; ++c) {
      const int k = tid + 256 * c;
      const bool kl = (k > 0), kr = (k + 1 < S_);
      float conv = bias + w01 * rowm[k] + w11 * rowc[k] + w21 * rowp[k];
      if (kl) conv += w00 * rowm[k - 1] + w10 * rowc[k - 1] + w20 * rowp[k - 1];
      if (kr) conv += w02 * rowm[k + 1] + w12 * rowc[k + 1] + w22 * rowp[k + 1];
      float val = rowc[k] + fmaxf(conv, 0.0f);
      if (mrow[k] == 0) val = -1e9f;
      vout[c] = val;
    }
    __syncthreads();                             // all reads of slab[r] done
#pragma unroll
    for (int c = 0; c < 8; ++c)
      slab[(size_t)r * PITCH + tid + 256 * c] = vout[c];
    // iter r+1 reads slab[r+1..r+3], disjoint from these writes -> no sync needed here
  }
  __syncthreads();

  // ---- Phase C: softmax, one wave per row (rows wave and wave+8) --------
  for (int rr = 0; rr < 2; ++rr) {
    float* p = slab + (size_t)(wave + rr * 8) * PITCH;
    float mx = -3.0e38f;
    for (int i = lane; i < S_; i += 32) mx = fmaxf(mx, p[i]);
#pragma unroll
    for (int off = 16; off > 0; off >>= 1) mx = fmaxf(mx, __shfl_xor(mx, off, 32));
    float sum = 0.0f;
    for (int i = lane; i < S_; i += 32) {
      float e = __expf(p[i] - mx);
      p[i] = e;
      sum += e;
    }
#pragma unroll
    for (int off = 16; off > 0; off >>= 1) sum += __shfl_xor(sum, off, 32);
    const float inv = 1.0f / sum;
    for (int i = lane; i < S_; i += 32) p[i] *= inv;
  }
  __syncthreads();

  // ---- Phase D: write atten tile (coalesced float4) ---------------------
  float* __restrict__ attn = out + CTX_ELEMS + ((size_t)b * S_ + q0) * (size_t)S_;
  for (int r = 0; r < 16; ++r) {
    const float* prow = slab + (size_t)r * PITCH;
    float* orow = attn + (size_t)r * S_;
    for (int c4 = tid; c4 < S_ / 4; c4 += 256) {
      *(float4*)(orow + c4 * 4) = *(const float4*)(prow + c4 * 4);
    }
  }

  // ---- Phase E: context(16x128) = P(16x2048) @ V(2048x128) via WMMA -----
  const float* __restrict__ vcol = V + (size_t)b * S_ * D_ + wave * 16 + m;
  const float* arow = slab + (size_t)m * PITCH;
  v8f acc = {0.f, 0.f, 0.f, 0.f, 0.f, 0.f, 0.f, 0.f};
  for (int kk = 0; kk < S_; kk += 4) {
    v2f a = *(const v2f*)(arow + kk + koff);     // ds_load_b64
    v2f bb;
    bb.x = vcol[(size_t)(kk + koff) * D_];
    bb.y = vcol[(size_t)(kk + koff + 1) * D_];
    acc = __builtin_amdgcn_wmma_f32_16x16x4_f32(false, a, false, bb,
                                                (short)0, acc, false, false);
  }
  float* __restrict__ ctx = out + ((size_t)b * S_ + q0) * (size_t)D_ + wave * 16 + m;
#pragma unroll
  for (int r = 0; r < 8; ++r) ctx[(size_t)(rbase + r) * D_] = acc[r];
}

// ---------------------------------------------------------------------------
extern "C" void kernel_launch(void* const* d_in, const int* in_sizes, int n_in,
                              void* d_out, int out_size, void* d_ws, size_t ws_size,
                              hipStream_t stream) {
  const float* Q  = (const float*)d_in[0];
  const float* K  = (const float*)d_in[1];
  const float* V  = (const float*)d_in[2];
  const int*   M  = (const int*)d_in[3];
  const float* W  = (const float*)d_in[4];
  const float* Bb = (const float*)d_in[5];
  float* out = (float*)d_out;

  dim3 grid(S_ / 16, B_);   // 2048 blocks of 256 threads (8 waves)
  fused_conv_attention_kernel<<<grid, 256, LDS_BYTES, stream>>>(Q, K, V, M, W, Bb, out);

  (void)in_sizes; (void)n_in; (void)out_size; (void)d_ws; (void)ws_size;
}